// DUVPTViT_35622458753376
// MI455X (gfx1250) — compile-verified
//
#include <hip/hip_runtime.h>
#include <stdint.h>

#define L_DIM 12
#define B_DIM 32
#define N_DIM 197
#define D_DIM 768
#define H_DIM 14
#define W_DIM 14
#define DC    256                 // D-chunk (768 = 3 * 256)
#define NCHUNK (D_DIM / DC)       // 3
#define ROWF  (W_DIM * DC)        // 3584 floats per row-chunk slab
#define NSLOT 4

// Generic pointers to LDS on amdgcn are {aperture_hi32, lds_offset32}:
// truncating to 32 bits yields the workgroup-relative LDS byte offset,
// which is exactly what GLOBAL_LOAD_ASYNC_TO_LDS consumes in its VDST VGPR.
__device__ __forceinline__ unsigned lds_byte_off(const void* p) {
    return (unsigned)(uintptr_t)p;
}

// Streaming read: each byte consumed exactly once (reuse happens in LDS) and
// the full 232MB stream exceeds the 192MB L2 -> non-temporal hint.
__device__ __forceinline__ void async_load_f4_nt(unsigned lds_off, const float* gaddr) {
    asm volatile("global_load_async_to_lds_b128 %0, %1, off th:TH_LOAD_NT"
                 :
                 : "v"(lds_off), "v"(gaddr)
                 : "memory");
}

__device__ __forceinline__ void wait_async0() {
#if __has_builtin(__builtin_amdgcn_s_wait_asynccnt)
    __builtin_amdgcn_s_wait_asynccnt(0);
#else
    asm volatile("s_wait_asynccnt 0" ::: "memory");
#endif
}

// One block per (l, b, dc): 12*32*3 = 1152 blocks, each streaming ~200KB.
// 1152 blocks x 14KB prefetch depth ~= 16MB in flight ~= HBM BW-delay product.
__global__ __launch_bounds__(256)
void duv_main_kernel(const float* __restrict__ feat,
                     const float* __restrict__ cmean,
                     const float* __restrict__ cstd,
                     float* __restrict__ ws_ap,   // [B*L][3] aleatoric partials
                     float* __restrict__ ws_e) {  // [B*L]   epistemic raw
    __shared__ float smem[NSLOT * ROWF];   // 57344 B ring buffer
    __shared__ float red_a[256];
    __shared__ float red_e[256];

    const int t   = threadIdx.x;
    const int idx = blockIdx.x;            // 0..1151
    const int dc  = idx % NCHUNK;
    const int rem = idx / NCHUNK;
    const int l   = rem / B_DIM;
    const int b   = rem % B_DIM;

    const size_t lstride = (size_t)B_DIM * N_DIM * D_DIM;
    const size_t bstride = (size_t)N_DIM * D_DIM;
    const float* base = feat + (size_t)l * lstride + (size_t)b * bstride;

    const unsigned smem_base = lds_byte_off(&smem[0]);

    float acc = 0.0f;

    // patch element (h, w, d) lives at base + (1 + h*W + w)*D + dc*DC + d
    const float* cbase = base + D_DIM + (size_t)dc * DC;

    // Preload rows 0 and 1 into slots 0, 1.
    for (int h0 = 0; h0 < 2; ++h0) {
        const float* rbase = cbase + (size_t)h0 * W_DIM * D_DIM;
        for (int e4 = t; e4 < ROWF / 4; e4 += 256) {
            const int e = e4 * 4;
            const int w = e >> 8;          // e / DC
            const int d = e & (DC - 1);    // e % DC
            async_load_f4_nt(smem_base + (unsigned)(h0 * ROWF + e) * 4u,
                             rbase + (size_t)w * D_DIM + d);
        }
    }
    wait_async0();
    __syncthreads();

    for (int h = 0; h < H_DIM; ++h) {
        // Prefetch row h+2 into slot (h+2)&3 (== (h-2)&3, free since the
        // barrier that ended iteration h-1).
        if (h + 2 < H_DIM) {
            const int slot = (h + 2) & 3;
            const float* rbase = cbase + (size_t)(h + 2) * W_DIM * D_DIM;
            for (int e4 = t; e4 < ROWF / 4; e4 += 256) {
                const int e = e4 * 4;
                const int w = e >> 8;
                const int d = e & (DC - 1);
                async_load_f4_nt(smem_base + (unsigned)(slot * ROWF + e) * 4u,
                                 rbase + (size_t)w * D_DIM + d);
            }
        }

        // Process row h: thread t owns lane d = t, walks all 14 w.
        const int sc = (h & 3) * ROWF;
        const int su = ((h + 3) & 3) * ROWF;   // row h-1
        const int sd = ((h + 1) & 3) * ROWF;   // row h+1
        #pragma unroll
        for (int w = 0; w < W_DIM; ++w) {
            const float c = smem[sc + w * DC + t];
            float s = 0.0f;
            int cnt = 0;
            if (h > 0)         { s += smem[su + w * DC + t];       ++cnt; }
            if (h < H_DIM - 1) { s += smem[sd + w * DC + t];       ++cnt; }
            if (w > 0)         { s += smem[sc + (w - 1) * DC + t]; ++cnt; }
            if (w < W_DIM - 1) { s += smem[sc + (w + 1) * DC + t]; ++cnt; }
            // cnt in {2,3,4}; branch-free reciprocal avoids v_div sequence
            const float invc = (cnt == 2) ? 0.5f
                             : ((cnt == 3) ? (1.0f / 3.0f) : 0.25f);
            const float diff = c - s * invc;
            acc = fmaf(diff, diff, acc);
        }

        wait_async0();
        __syncthreads();
    }

    // Epistemic: CLS token (n == 0), full 768-dim calibrated squared norm.
    // Only the dc==0 block of each (l,b) computes it (it is dc-independent).
    float acce = 0.0f;
    if (dc == 0) {
        #pragma unroll
        for (int k = 0; k < D_DIM / 256; ++k) {
            const int d = t + k * 256;
            const float v = (base[d] - cmean[(size_t)l * D_DIM + d]) /
                            (cstd[(size_t)l * D_DIM + d] + 1e-8f);
            acce = fmaf(v, v, acce);
        }
    }

    // Block reduction of both accumulators.
    red_a[t] = acc;
    red_e[t] = acce;
    __syncthreads();
    for (int s = 128; s > 0; s >>= 1) {
        if (t < s) {
            red_a[t] += red_a[t + s];
            red_e[t] += red_e[t + s];
        }
        __syncthreads();
    }
    if (t == 0) {
        ws_ap[(b * L_DIM + l) * NCHUNK + dc] = red_a[0];
        if (dc == 0) ws_e[b * L_DIM + l] = red_e[0];
    }
}

__global__ __launch_bounds__(32)
void duv_norm_kernel(const float* __restrict__ ws_ap,
                     const float* __restrict__ ws_e,
                     float* __restrict__ out) {
    const int b = blockIdx.x;
    const int l = threadIdx.x;
    if (l >= L_DIM) return;
    float my_a = 0.0f;
    const float my_e = ws_e[b * L_DIM + l];
    float ma = 0.0f, me = 0.0f;   // values are sums of squares -> >= 0
    for (int j = 0; j < L_DIM; ++j) {
        const int p = (b * L_DIM + j) * NCHUNK;
        const float a = (ws_ap[p] + ws_ap[p + 1] + ws_ap[p + 2]) *
                        (1.0f / (float)(H_DIM * W_DIM));
        const float e = ws_e[b * L_DIM + j];
        if (j == l) my_a = a;
        ma = fmaxf(ma, a);
        me = fmaxf(me, e);
    }
    out[b * L_DIM + l]                 = my_a / (ma + 1e-8f);
    out[B_DIM * L_DIM + b * L_DIM + l] = my_e / (me + 1e-8f);
}

extern "C" void kernel_launch(void* const* d_in, const int* in_sizes, int n_in,
                              void* d_out, int out_size, void* d_ws, size_t ws_size,
                              hipStream_t stream) {
    const float* feat = (const float*)d_in[0];   // [12, 32, 197, 768] f32
    const float* cm   = (const float*)d_in[1];   // [12, 768] f32
    const float* cs   = (const float*)d_in[2];   // [12, 768] f32

    float* ws_ap = (float*)d_ws;                          // [384*3] partials
    float* ws_e  = ws_ap + (size_t)L_DIM * B_DIM * NCHUNK; // [384]

    duv_main_kernel<<<L_DIM * B_DIM * NCHUNK, 256, 0, stream>>>(feat, cm, cs, ws_ap, ws_e);
    duv_norm_kernel<<<B_DIM, 32, 0, stream>>>(ws_ap, ws_e, (float*)d_out);
}